// Sampler_6107443495068
// MI455X (gfx1250) — compile-verified
//
#include <hip/hip_runtime.h>
#include <math.h>
#include <stdint.h>

// ---------------------------------------------------------------------------
// Sampler for B=128 rows x V=128000 f32 logits (see reference).
//
// Kernel 1 (640 blocks = 128 rows x 5 slices, 256 thr = 8 wave32):
//   single HBM pass over its 25600-float slice through an 8-deep ring of LDS
//   buffers fed by CDNA5 async global->LDS loads (ASYNCcnt). Per element:
//     - online softmax stats (1-exp update) of raw logits
//     - per-thread top-20 of raw logits -> block-merged slice top-20
//     - penalty pipeline (stop mask, repetition/freq/presence, temperature)
//       via slice-local LDS bitmaps + count hash
//     - per-thread register top-4 of penalized values
//   The 256x4 union contains the slice top-64 whp; a tiny histogram over the
//   union (1024 adds, no hot-bin serialization) picks a threshold, the union
//   is gathered and wave0 extracts the slice top-64 -> device buffer.
//
// Kernel 2 (128 blocks): merges 5 slices/row: log-softmax normalizer, row
//   top-20 logprobs, top-128 candidates (head = penalized argmax = greedy),
//   then top-k kth value (+ties), survivor softmax, ascending top-p cumsum
//   mask, gumbel argmax, temperature<eps -> greedy.
// ---------------------------------------------------------------------------

#define BDIM 256
#define NWAVE 8
static constexpr int   B       = 128;
static constexpr int   V       = 128000;
static constexpr int   P       = 1024;
static constexpr int   O       = 256;
static constexpr int   NL      = 20;            // max_num_logprobs
static constexpr int   SLICES  = 5;             // blocks per row
static constexpr int   SLICE   = V / SLICES;    // 25600 floats
static constexpr int   SVW     = SLICE / 32;    // 800 bitmap words
static constexpr int   HBINS   = 2048;          // histogram bins (key >> 21)
static constexpr int   HSH     = 512;           // hash slots for output counts
static constexpr int   TPT     = 4;             // per-thread top-4 of penalized
static constexpr int   NU      = BDIM * TPT;    // union size = 1024
static constexpr int   CCAP    = 256;           // gathered candidate cap
static constexpr int   NCAND   = 64;            // candidates kept per slice
static constexpr int   CHUNK   = 1024;          // floats per streamed chunk
static constexpr int   NCHUNK  = SLICE / CHUNK; // 25
static constexpr int   DEPTH   = 8;             // async ring depth
static constexpr int   LSORT   = 128;           // merged sorted candidate cap
static constexpr float NEG_INF = -3.4028234663852886e38f;
static constexpr float EPSV    = 1e-5f;

// ---- inter-kernel per-slice partials (fully rewritten every launch) --------
struct alignas(16) SliceRes {
  float m, d;                       // online softmax stats of raw logits
  float t20v[NL]; int t20i[NL];     // slice top-20 of raw logits (descending)
  float cv[NCAND]; int ci[NCAND];   // slice top-64 penalized candidates
};
__device__ SliceRes g_slices[B * SLICES];

// ---------------- CDNA5 async global->LDS data mover ------------------------
// GV mode: VDST = LDS byte address (vgpr), VADDR = 64-bit global address.
// Tracked with ASYNCcnt (CDNA5 ISA).
__device__ __forceinline__ void async_b128_to_lds(uint32_t lds_byte, const float* g) {
  unsigned long long ga = (unsigned long long)(uintptr_t)g;
  asm volatile("global_load_async_to_lds_b128 %0, %1, off"
               :: "v"(lds_byte), "v"(ga) : "memory");
}
__device__ __forceinline__ void wait_async_le(int n) {
  switch (n) {
    case 0:  asm volatile("s_wait_asynccnt 0" ::: "memory"); break;
    case 1:  asm volatile("s_wait_asynccnt 1" ::: "memory"); break;
    case 2:  asm volatile("s_wait_asynccnt 2" ::: "memory"); break;
    case 3:  asm volatile("s_wait_asynccnt 3" ::: "memory"); break;
    case 4:  asm volatile("s_wait_asynccnt 4" ::: "memory"); break;
    case 5:  asm volatile("s_wait_asynccnt 5" ::: "memory"); break;
    case 6:  asm volatile("s_wait_asynccnt 6" ::: "memory"); break;
    default: asm volatile("s_wait_asynccnt 7" ::: "memory"); break;
  }
}

// ---------------- helpers ---------------------------------------------------
__device__ __forceinline__ uint32_t ordf(float f) {            // order-preserving
  uint32_t u = __float_as_uint(f);
  return (u & 0x80000000u) ? ~u : (u | 0x80000000u);
}
__device__ __forceinline__ float unordf(uint32_t u) {
  uint32_t fb = (u & 0x80000000u) ? (u ^ 0x80000000u) : ~u;
  return __uint_as_float(fb);
}
__device__ __forceinline__ bool kbetter(float v, int i, float bv, int bi) {
  return (v > bv) || (v == bv && i < bi);        // lowest-index tie-break
}
__device__ __forceinline__ unsigned long long wave_max_u64(unsigned long long k) {
#pragma unroll
  for (int off = 16; off > 0; off >>= 1) {
    unsigned long long o = __shfl_xor(k, off, 32);
    if (o > k) k = o;
  }
  return k;
}
__device__ __forceinline__ unsigned long long block_max_u64(
    unsigned long long k, volatile unsigned long long* scr, int tid) {
  k = wave_max_u64(k);
  if ((tid & 31) == 0) scr[tid >> 5] = k;
  __syncthreads();
  unsigned long long best = scr[0];
#pragma unroll
  for (int i = 1; i < NWAVE; ++i) { unsigned long long o = scr[i]; if (o > best) best = o; }
  __syncthreads();
  return best;
}
// one wave repeatedly extracts the max (value, lowest-index tie-break)
__device__ __forceinline__ void wave_select_topn(volatile float* vals, volatile int* idxs,
                                                 int n, float* outv, int* outi,
                                                 int rounds, int lane) {
  for (int r = 0; r < rounds; ++r) {
    unsigned long long key = 0ull;
    for (int i = lane; i < n; i += 32) {
      unsigned long long k =
          ((unsigned long long)ordf(vals[i]) << 32) | (uint32_t)~(uint32_t)idxs[i];
      if (k > key) key = k;
    }
    key = wave_max_u64(key);
    if (lane == 0) {
      outv[r] = unordf((uint32_t)(key >> 32));
      outi[r] = (int)~(uint32_t)(key & 0xFFFFFFFFull);
    }
    for (int i = lane; i < n; i += 32) {
      unsigned long long k =
          ((unsigned long long)ordf(vals[i]) << 32) | (uint32_t)~(uint32_t)idxs[i];
      if (k == key) { vals[i] = NEG_INF; idxs[i] = 0x7FFFFFFF; }
    }
  }
}
// deterministic hash RNG for the gumbel noise (stand-in for jax threefry)
__device__ __forceinline__ float rng_uniform(uint32_t a, uint32_t b) {
  uint32_t s = a * 0x9E3779B9u + b * 0x85EBCA6Bu + 0x27220A95u;
  s ^= s >> 16; s *= 0x7FEB352Du; s ^= s >> 15; s *= 0x846CA68Bu; s ^= s >> 16;
  float f = (float)(s >> 8) * (1.0f / 16777216.0f);
  return fminf(fmaxf(f, 1e-10f), 0.99999994f);
}

struct RowParams {
  float rep, invrep, freq, pres, invtemp;
  int st0, st1, st2, st3;
  int penalize;   // output_len < min_tokens
};

__device__ __forceinline__ int hash_lookup(const int* hkey, const int* hcnt, int tok) {
  uint32_t h = ((uint32_t)tok * 2654435761u) & (HSH - 1);
  for (;;) {
    int k = hkey[h];
    if (k == tok) return hcnt[h];
    if (k == -1) return 0;
    h = (h + 1) & (HSH - 1);
  }
}

__device__ __forceinline__ float apply_pipeline(float x, int v, const RowParams& rp,
                                                uint32_t pw, uint32_t ow,
                                                const int* hkey, const int* hcnt) {
  if (rp.penalize && (v == rp.st0 || v == rp.st1 || v == rp.st2 || v == rp.st3))
    x = NEG_INF;
  uint32_t bit = 1u << (v & 31);
  bool ob = (ow & bit) != 0;
  bool pb = (pw & bit) != 0;
  if (ob | pb) x = (x > 0.0f) ? x * rp.invrep : x * rp.rep;
  if (ob) {
    float cnt = (float)hash_lookup(hkey, hcnt, v);
    x -= rp.freq * cnt + rp.pres;
  }
  return x * rp.invtemp;
}

// ---------------------------------------------------------------------------
// Kernel 1: per-(row, slice) single streaming pass
// ---------------------------------------------------------------------------
__global__ __launch_bounds__(BDIM)
void sampler_slice_kernel(const float* __restrict__ logits,
                          const float* __restrict__ temperature,
                          const float* __restrict__ presence,
                          const float* __restrict__ frequency,
                          const float* __restrict__ repetition,
                          const int*   __restrict__ prompt_ids,
                          const int*   __restrict__ out_tok,
                          const int*   __restrict__ out_lens,
                          const int*   __restrict__ stop_ids,
                          const int*   __restrict__ min_tok) {
  __shared__ uint32_t sh_pbits[SVW];
  __shared__ uint32_t sh_obits[SVW];
  __shared__ int      sh_hkey[HSH];
  __shared__ int      sh_hcnt[HSH];
  __shared__ __align__(16) float sh_stage[DEPTH][CHUNK];   // 32 KB async ring
  __shared__ float    sh_uv[NU];                 // union of per-thread top-4
  __shared__ int      sh_ui[NU];
  __shared__ uint32_t sh_hist[HBINS];
  __shared__ uint32_t sh_gsum[BDIM];
  __shared__ float    sh_cval[CCAP];
  __shared__ int      sh_cidx[CCAP];
  __shared__ unsigned long long sh_red64[NWAVE];
  __shared__ float    sh_redm[NWAVE], sh_redd[NWAVE];
  __shared__ int      sh_cn;
  __shared__ uint32_t sh_uthr;
  __shared__ float    sh_t20v[NL];
  __shared__ int      sh_t20i[NL];

  const int s    = blockIdx.x;
  const int b    = blockIdx.y;
  const int tid  = threadIdx.x;
  const int wid  = tid >> 5;
  const int lane = tid & 31;
  const int lo   = s * SLICE;
  const float* rowslice = logits + (size_t)b * V + lo;
  SliceRes* gs = &g_slices[b * SLICES + s];

  // ---- zero/init LDS --------------------------------------------------------
  for (int i = tid; i < SVW; i += BDIM) { sh_pbits[i] = 0u; sh_obits[i] = 0u; }
  for (int i = tid; i < HSH; i += BDIM) { sh_hkey[i] = -1; sh_hcnt[i] = 0; }
  for (int i = tid; i < HBINS; i += BDIM) sh_hist[i] = 0u;
  if (tid == 0) sh_cn = 0;
  __syncthreads();

  // ---- build slice-local bitmaps + count hash (ds atomics) ------------------
  for (int j = tid; j < P; j += BDIM) {
    int tok = prompt_ids[b * P + j];
    if (tok >= lo && tok < lo + SLICE)
      atomicOr(&sh_pbits[(tok - lo) >> 5], 1u << (tok & 31));
  }
  const int olen = out_lens[b];
  for (int j = tid; j < O; j += BDIM) {
    if (j < olen) {
      int tok = out_tok[b * O + j];
      if (tok >= lo && tok < lo + SLICE) {
        atomicOr(&sh_obits[(tok - lo) >> 5], 1u << (tok & 31));
        uint32_t h = ((uint32_t)tok * 2654435761u) & (HSH - 1);
        for (;;) {
          int prev = atomicCAS(&sh_hkey[h], -1, tok);
          if (prev == -1 || prev == tok) { atomicAdd(&sh_hcnt[h], 1); break; }
          h = (h + 1) & (HSH - 1);
        }
      }
    }
  }
  __syncthreads();

  // ---- per-row params -------------------------------------------------------
  RowParams rp;
  rp.rep    = repetition[b];
  rp.invrep = 1.0f / rp.rep;
  rp.freq   = frequency[b];
  rp.pres   = presence[b];
  float temp = temperature[b];
  rp.invtemp = 1.0f / ((temp < EPSV) ? 1.0f : temp);
  rp.st0 = stop_ids[b * 4 + 0]; rp.st1 = stop_ids[b * 4 + 1];
  rp.st2 = stop_ids[b * 4 + 2]; rp.st3 = stop_ids[b * 4 + 3];
  rp.penalize = (olen < min_tok[b]) ? 1 : 0;

  // ---- Pass A: async-streamed HBM pass, 8-deep ring, no loop barriers -------
  // Each wave owns a 512B slice of each ring buffer and only reads what it
  // loaded, so waves run fully decoupled; ASYNCcnt tracks its own chunks.
  const uint32_t stage_base = (uint32_t)(uintptr_t)(&sh_stage[0][0]);
  const uint32_t my_slot    = (uint32_t)(wid * 512 + lane * 16);

  float mraw = NEG_INF, draw = 0.0f;
  float t20v[NL]; int t20i[NL];
#pragma unroll
  for (int j = 0; j < NL; ++j) { t20v[j] = NEG_INF; t20i[j] = 0x7FFFFFFF; }
  // per-thread top-4 of penalized values (explicit scalars: stays in VGPRs)
  float a0 = NEG_INF, a1 = NEG_INF, a2 = NEG_INF, a3 = NEG_INF;
  int   i0 = 0x7FFFFFFF, i1 = 0x7FFFFFFF, i2 = 0x7FFFFFFF, i3 = 0x7FFFFFFF;

  int ni = 0;                                   // next chunk to issue
  for (; ni < DEPTH && ni < NCHUNK; ++ni)
    async_b128_to_lds(stage_base + (uint32_t)((ni & (DEPTH - 1)) * (CHUNK * 4)) + my_slot,
                      rowslice + (size_t)ni * CHUNK + wid * 128 + lane * 4);

  for (int c = 0; c < NCHUNK; ++c) {
    wait_async_le(ni - c - 1);                  // chunk c has landed in LDS
    const float4 x4 = *(const float4*)&sh_stage[c & (DEPTH - 1)][wid * 128 + lane * 4];
    const int vloc0 = c * CHUNK + wid * 128 + lane * 4;   // 4-aligned -> one word
    const uint32_t pw = sh_pbits[vloc0 >> 5];
    const uint32_t ow = sh_obits[vloc0 >> 5];
    float xs[4] = {x4.x, x4.y, x4.z, x4.w};
#pragma unroll
    for (int j = 0; j < 4; ++j) {
      float x = xs[j]; int v = lo + vloc0 + j;
      // raw log-softmax stats (online, 1 exp on the common path)
      if (x <= mraw) {
        draw += __expf(x - mraw);
      } else {
        draw = draw * __expf(mraw - x) + 1.0f;
        mraw = x;
      }
      // per-thread top-20 of raw logits (bubble insert, mostly skipped)
      if (x > t20v[NL - 1]) {
        float cvv = x; int cii = v;
#pragma unroll
        for (int q = 0; q < NL; ++q) {
          bool sw = kbetter(cvv, cii, t20v[q], t20i[q]);
          float fv = sw ? t20v[q] : cvv; int fi = sw ? t20i[q] : cii;
          t20v[q] = sw ? cvv : t20v[q]; t20i[q] = sw ? cii : t20i[q];
          cvv = fv; cii = fi;
        }
      }
      // penalized value -> per-thread top-4 (rare insert)
      float sv = apply_pipeline(x, v, rp, pw, ow, sh_hkey, sh_hcnt);
      if (kbetter(sv, v, a3, i3)) {
        float cv = sv; int ci = v;
        if (kbetter(cv, ci, a0, i0)) { float t = a0; int ti = i0; a0 = cv; i0 = ci; cv = t; ci = ti; }
        if (kbetter(cv, ci, a1, i1)) { float t = a1; int ti = i1; a1 = cv; i1 = ci; cv = t; ci = ti; }
        if (kbetter(cv, ci, a2, i2)) { float t = a2; int ti = i2; a2 = cv; i2 = ci; cv = t; ci = ti; }
        a3 = cv; i3 = ci;
      }
    }
    if (ni < NCHUNK) {                          // refill ring
      async_b128_to_lds(stage_base + (uint32_t)((ni & (DEPTH - 1)) * (CHUNK * 4)) + my_slot,
                        rowslice + (size_t)ni * CHUNK + wid * 128 + lane * 4);
      ++ni;
    }
  }

  // ---- publish per-thread top-4 union + histogram it (1024 adds only) -------
  sh_uv[tid * TPT + 0] = a0; sh_ui[tid * TPT + 0] = i0;
  sh_uv[tid * TPT + 1] = a1; sh_ui[tid * TPT + 1] = i1;
  sh_uv[tid * TPT + 2] = a2; sh_ui[tid * TPT + 2] = i2;
  sh_uv[tid * TPT + 3] = a3; sh_ui[tid * TPT + 3] = i3;
  atomicAdd(&sh_hist[ordf(a0) >> 21], 1u);
  atomicAdd(&sh_hist[ordf(a1) >> 21], 1u);
  atomicAdd(&sh_hist[ordf(a2) >> 21], 1u);
  atomicAdd(&sh_hist[ordf(a3) >> 21], 1u);

  // ---- reduce raw softmax stats (wave32 shuffles + LDS) ---------------------
  {
    float m = mraw, d = draw;
#pragma unroll
    for (int off = 16; off > 0; off >>= 1) {
      float mo = __shfl_xor(m, off, 32);
      float doo = __shfl_xor(d, off, 32);
      float nm = fmaxf(m, mo);
      d = d * __expf(m - nm) + doo * __expf(mo - nm);
      m = nm;
    }
    if (lane == 0) { sh_redm[wid] = m; sh_redd[wid] = d; }
    __syncthreads();
    if (tid == 0) {
      float M = sh_redm[0], D = sh_redd[0];
#pragma unroll
      for (int i = 1; i < NWAVE; ++i) {
        float mo = sh_redm[i], doo = sh_redd[i];
        float nm = fmaxf(M, mo);
        D = D * __expf(M - nm) + doo * __expf(mo - nm);
        M = nm;
      }
      gs->m = M; gs->d = D;
    }
    __syncthreads();
  }

  // ---- merge per-thread top-20 lists (20 rounds of block argmax) ------------
  {
    int hp = 0;
    for (int r = 0; r < NL; ++r) {
      unsigned long long key = 0ull;
      if (hp < NL)
        key = ((unsigned long long)ordf(t20v[hp]) << 32) | (uint32_t)~(uint32_t)t20i[hp];
      unsigned long long best = block_max_u64(key, sh_red64, tid);
      if (key == best && key != 0ull) {
        sh_t20v[r] = t20v[hp]; sh_t20i[r] = t20i[hp]; ++hp;
      }
    }
    __syncthreads();
    if (tid < NL) { gs->t20v[tid] = sh_t20v[tid]; gs->t20i[tid] = sh_t20i[tid]; }
  }

  // ---- union histogram suffix-select: threshold covering top-64 -------------
  {
    uint32_t gsum = 0;
#pragma unroll
    for (int i = 0; i < HBINS / BDIM; ++i) gsum += sh_hist[tid * (HBINS / BDIM) + i];
    sh_gsum[tid] = gsum;
    __syncthreads();
    if (tid == 0) {
      const uint32_t need = NCAND;
      uint32_t acc = 0; int g = BDIM - 1;
      for (; g > 0; --g) { acc += sh_gsum[g]; if (acc >= need) break; }
      if (acc < need) acc += sh_gsum[0];
      uint32_t above = acc - sh_gsum[g];
      int bsel = g * (HBINS / BDIM);
      uint32_t acc2 = above;
      for (int bb = g * (HBINS / BDIM) + (HBINS / BDIM) - 1; bb >= g * (HBINS / BDIM); --bb) {
        acc2 += sh_hist[bb];
        if (acc2 >= need) { bsel = bb; break; }
      }
      sh_uthr = (uint32_t)bsel << 21;
    }
    __syncthreads();
  }
  const uint32_t uthr = sh_uthr;

  // ---- gather union entries above threshold ---------------------------------
#pragma unroll
  for (int q = 0; q < TPT; ++q) {
    float sv = sh_uv[tid * TPT + q];
    int   vi = sh_ui[tid * TPT + q];
    if (ordf(sv) >= uthr) {
      int slot = atomicAdd(&sh_cn, 1);
      if (slot < CCAP) { sh_cval[slot] = sv; sh_cidx[slot] = vi; }
    }
  }
  __syncthreads();

  // ---- wave 0: top-64 of candidates -> device buffer ------------------------
  const int C  = (sh_cn < CCAP) ? sh_cn : CCAP;
  const int L1 = (C < NCAND) ? C : NCAND;
  if (wid == 0) {
    wave_select_topn(sh_cval, sh_cidx, C, gs->cv, gs->ci, L1, lane);
    for (int r = L1 + lane; r < NCAND; r += 32) { gs->cv[r] = NEG_INF; gs->ci[r] = 0x7FFFFFFF; }
  }
}

// ---------------------------------------------------------------------------
// Kernel 2: per-row merge + sampling tail
// ---------------------------------------------------------------------------
__global__ __launch_bounds__(BDIM)
void sampler_merge_kernel(const float* __restrict__ temperature,
                          const float* __restrict__ top_p,
                          const int*   __restrict__ top_k,
                          int*   __restrict__ out_sampled,
                          float* __restrict__ out_lp,
                          int*   __restrict__ out_li) {
  __shared__ float sh_cv[SLICES * NCAND];
  __shared__ int   sh_ci[SLICES * NCAND];
  __shared__ float sh_tv[SLICES * NL];
  __shared__ int   sh_ti[SLICES * NL];
  __shared__ float sh_sv[LSORT];
  __shared__ int   sh_si[LSORT];
  __shared__ float sh_ov[NL];
  __shared__ int   sh_oi[NL];
  __shared__ float sh_logZ;

  const int b = blockIdx.x, tid = threadIdx.x, wid = tid >> 5, lane = tid & 31;
  const SliceRes* gs = &g_slices[b * SLICES];

  for (int i = tid; i < SLICES * NCAND; i += BDIM) {
    sh_cv[i] = gs[i / NCAND].cv[i % NCAND];
    sh_ci[i] = gs[i / NCAND].ci[i % NCAND];
  }
  for (int i = tid; i < SLICES * NL; i += BDIM) {
    sh_tv[i] = gs[i / NL].t20v[i % NL];
    sh_ti[i] = gs[i / NL].t20i[i % NL];
  }
  if (tid == 0) {
    float M = gs[0].m, D = gs[0].d;
#pragma unroll
    for (int i = 1; i < SLICES; ++i) {
      float mo = gs[i].m, doo = gs[i].d;
      float nm = fmaxf(M, mo);
      D = D * __expf(M - nm) + doo * __expf(mo - nm);
      M = nm;
    }
    sh_logZ = M + __logf(D);
  }
  __syncthreads();

  // wave 0: row top-20; wave 1: row top-128 candidates (independent LDS)
  if (wid == 0) wave_select_topn(sh_tv, sh_ti, SLICES * NL, sh_ov, sh_oi, NL, lane);
  if (wid == 1) wave_select_topn(sh_cv, sh_ci, SLICES * NCAND, sh_sv, sh_si, LSORT, lane);
  __syncthreads();

  if (tid == 0) {
    // top-20 logprobs = raw logit - logZ
    float logZ = sh_logZ;
    for (int r = 0; r < NL; ++r) {
      out_lp[b * NL + r] = sh_ov[r] - logZ;
      out_li[b * NL + r] = sh_oi[r];
    }
    // greedy argmax of penalized logits = head of merged candidates
    const int greedy = sh_si[0];
    // count real (non-pad) candidates
    int L = 0;
    while (L < LSORT && sh_si[L] != 0x7FFFFFFF) ++L;
    float temp = temperature[b];
    int sampled;
    if (L <= 0) {
      sampled = greedy;
    } else {
      int kk = top_k[b]; kk = (kk < 1) ? 1 : ((kk > V) ? V : kk);
      int kidx = (kk - 1 < L) ? (kk - 1) : (L - 1);
      float kth = sh_sv[kidx];
      int Ns = kidx + 1;
      while (Ns < L && sh_sv[Ns] >= kth) ++Ns;        // include kth-value ties
      float m = sh_sv[0];
      float Z = 0.0f;
      for (int j = 0; j < Ns; ++j) Z += __expf(sh_sv[j] - m);
      float invZ = 1.0f / Z;
      float keep_thr = 1.0f - top_p[b];
      float best = NEG_INF; int bestIdx = sh_si[0];
      float cum = 0.0f;
      for (int j = Ns - 1; j >= 0; --j) {             // ascending order
        cum += __expf(sh_sv[j] - m) * invZ;
        bool masked = (j != 0) && (cum <= keep_thr);
        if (!masked) {
          float u = rng_uniform((uint32_t)b, (uint32_t)sh_si[j]);
          float g = -__logf(-__logf(u));              // gumbel
          float sc = sh_sv[j] + g;
          if (sc > best) { best = sc; bestIdx = sh_si[j]; }
        }
      }
      sampled = (temp < EPSV) ? greedy : bestIdx;
    }
    out_sampled[b] = sampled;
  }
}

// ---------------------------------------------------------------------------
extern "C" void kernel_launch(void* const* d_in, const int* in_sizes, int n_in,
                              void* d_out, int out_size, void* d_ws, size_t ws_size,
                              hipStream_t stream) {
  (void)in_sizes; (void)n_in; (void)out_size; (void)d_ws; (void)ws_size;
  const float* logits      = (const float*)d_in[0];
  const float* temperature = (const float*)d_in[1];
  const float* presence    = (const float*)d_in[2];
  const float* frequency   = (const float*)d_in[3];
  const float* repetition  = (const float*)d_in[4];
  const float* topp        = (const float*)d_in[5];
  const int*   prompt_ids  = (const int*)d_in[6];
  const int*   out_tok     = (const int*)d_in[7];
  const int*   out_lens    = (const int*)d_in[8];
  const int*   stop_ids    = (const int*)d_in[9];
  const int*   min_tok     = (const int*)d_in[10];
  const int*   topk        = (const int*)d_in[11];

  // tuple output flat: sampled[B] (int bits), topk_logprobs[B*20], topk_indices[B*20]
  int*   out_sampled = (int*)d_out;
  float* out_lp      = (float*)d_out + B;
  int*   out_li      = (int*)d_out + B + B * NL;

  hipLaunchKernelGGL(sampler_slice_kernel, dim3(SLICES, B), dim3(BDIM), 0, stream,
                     logits, temperature, presence, frequency, repetition,
                     prompt_ids, out_tok, out_lens, stop_ids, min_tok);
  hipLaunchKernelGGL(sampler_merge_kernel, dim3(B), dim3(BDIM), 0, stream,
                     temperature, topp, topk, out_sampled, out_lp, out_li);
}